// Encoder_LSTM_1_33062658245301
// MI455X (gfx1250) — compile-verified
//
#include <hip/hip_runtime.h>
#include <hip/hip_bf16.h>

#define VOCAB   128
#define EMBED   256
#define HIDDEN  512
#define BATCH   512
#define SEQT    64
#define GATES   (4*HIDDEN)   // 2048
#define OUT2H   (2*HIDDEN)   // 1024
#define KCHUNK  64           // K rows staged per async chunk in the step kernel

typedef float v2f __attribute__((ext_vector_type(2)));
typedef float v8f __attribute__((ext_vector_type(8)));

__device__ __forceinline__ float sigmoid_f(float x) {
  x = fminf(fmaxf(x, -30.0f), 30.0f);
  return 1.0f / (1.0f + __expf(-x));
}
__device__ __forceinline__ float tanh_f(float x) {
  x = fminf(fmaxf(x, -15.0f), 15.0f);
  float e = __expf(2.0f * x);
  return (e - 1.0f) / (e + 1.0f);
}

__device__ __forceinline__ v8f wmma_f32x4(v2f a, v2f b, v8f c) {
  // D = A(16x4 f32) * B(4x16 f32) + C(16x16 f32)  -> v_wmma_f32_16x16x4_f32
  return __builtin_amdgcn_wmma_f32_16x16x4_f32(false, a, false, b, (short)0, c,
                                               false, false);
}

// Async global->LDS copy, 16 bytes per lane per issue (ASYNCcnt tracked).
// lds_off is the 32-bit LDS byte address (low 32 bits of the generic pointer).
__device__ __forceinline__ void async_copy_b128(unsigned int lds_off,
                                                const float* gptr) {
  asm volatile("global_load_async_to_lds_b128 %0, %1, off"
               :: "v"(lds_off), "v"(gptr) : "memory");
}
__device__ __forceinline__ void wait_async0() {
  asm volatile("s_wait_asynccnt 0x0" ::: "memory");
}
__device__ __forceinline__ unsigned int lds_addr32(const void* p) {
  return (unsigned int)(unsigned long long)p;   // AS3 offset = low 32 bits
}

// ---------------------------------------------------------------------------
// Init: zero LSTM state (ping-pong h buffers + c) and emit src_mask output.
// ---------------------------------------------------------------------------
__global__ __launch_bounds__(256)
void init_kernel(const int* __restrict__ src_ids, float* __restrict__ mask_out,
                 float* __restrict__ hf0, float* __restrict__ hf1,
                 float* __restrict__ cf,
                 float* __restrict__ hb0, float* __restrict__ hb1,
                 float* __restrict__ cb)
{
  int i = blockIdx.x * blockDim.x + threadIdx.x;
  if (i < BATCH * HIDDEN) {
    hf0[i] = 0.0f; hf1[i] = 0.0f; cf[i] = 0.0f;
    hb0[i] = 0.0f; hb1[i] = 0.0f; cb[i] = 0.0f;
  }
  if (i < BATCH * SEQT) {
    mask_out[i] = (src_ids[i] != 0) ? 1.0f : 0.0f;
  }
}

// ---------------------------------------------------------------------------
// Fused embedding-gather + input projection:
//   xz[m, g] = embed[src_ids[m]] @ K + bias     (m = b*T + t), both directions
// The 256x16 B-panel for this block's column tile is staged once into LDS
// with async global->LDS b128 copies and shared by all 4 waves.
// ---------------------------------------------------------------------------
__global__ __launch_bounds__(128)
void xz_gemm_kernel(const int* __restrict__ src_ids,
                    const float* __restrict__ embed,
                    const float* __restrict__ Kf, const float* __restrict__ bf,
                    const float* __restrict__ Kb, const float* __restrict__ bb,
                    float* __restrict__ xz_f, float* __restrict__ xz_b)
{
  __shared__ float Bs[EMBED * 16];   // 16 KB K-panel

  const int tid   = threadIdx.x;
  const int lane  = tid & 31;
  const int wave  = tid >> 5;
  const int lm    = lane & 15;
  const int hi    = lane >> 4;        // 0 or 1
  const int khalf = hi * 2;           // K offset 0 or 2 within the 4-chunk

  const int colBase = blockIdx.x * 16;
  const int mBase   = (blockIdx.y * 4 + wave) * 16;
  const int dir     = blockIdx.z;

  const float* Kw   = dir ? Kb : Kf;
  const float* bias = dir ? bb : bf;
  float*       xz   = dir ? xz_b : xz_f;

  const int col = colBase + lm;

  // ---- async stage of the B panel: EMBED rows x 16 cols (16B aligned) ----
  {
    const float*       gbase = Kw + colBase;
    const unsigned int lds0  = lds_addr32(Bs);
    #pragma unroll
    for (int i = tid; i < (EMBED * 16) / 4; i += 128) {   // 8 iters/thread
      const int e   = i * 4;
      const int row = e >> 4;
      const int c   = e & 15;
      async_copy_b128(lds0 + (unsigned)e * 4u, gbase + (long)row * GATES + c);
    }
  }

  // Embedding row for this lane's A-fragment rows (lanes 0-15 / 16-31 share M)
  const int    id   = src_ids[mBase + lm];
  const float* Arow = embed + (long)id * EMBED;

  wait_async0();
  __syncthreads();

  v8f acc = {};
  #pragma unroll 4
  for (int k = 0; k < EMBED; k += 4) {
    const int k0 = k + khalf;
    v2f a; a.x = Arow[k0];            a.y = Arow[k0 + 1];
    v2f b; b.x = Bs[k0 * 16 + lm];    b.y = Bs[(k0 + 1) * 16 + lm];
    acc = wmma_f32x4(a, b, acc);
  }

  const float bv = bias[col];
  #pragma unroll
  for (int v = 0; v < 8; ++v) {
    const int row = mBase + v + hi * 8;     // C/D layout: VGPR v -> M = v (+8 hi)
    xz[(long)row * GATES + col] = acc[v] + bv;
  }
}

// ---------------------------------------------------------------------------
// One LSTM timestep, both directions (grid.z). Per wave: one 16(batch)x16(h)
// tile with four gate accumulators sharing one h_prev A-fragment. The R panel
// (4 gates x 512 rows x 16 cols = 128 KB) is streamed through LDS in 64-row
// chunks, double-buffered with async global->LDS copies overlapping the WMMA
// compute. h is ping-pong buffered across launches (read h_in, write h_out).
// ---------------------------------------------------------------------------
__global__ __launch_bounds__(128)
void lstm_step_kernel(const float* __restrict__ xz_f, const float* __restrict__ xz_b,
                      const float* __restrict__ Rf, const float* __restrict__ Rb,
                      const int* __restrict__ src_ids,
                      const float* __restrict__ hf_in, float* __restrict__ hf_out,
                      float* __restrict__ cf,
                      const float* __restrict__ hb_in, float* __restrict__ hb_out,
                      float* __restrict__ cb,
                      float* __restrict__ seq_out, int step)
{
  __shared__ float Bs[2][4 * KCHUNK * 16];   // 2 x 16 KB double buffer

  const int tid  = threadIdx.x;
  const int lane = tid & 31;
  const int wave = tid >> 5;
  const int lm   = lane & 15;
  const int hi   = lane >> 4;

  const int dir = blockIdx.z;
  const int t   = dir ? (SEQT - 1 - step) : step;

  const float* R     = dir ? Rb    : Rf;
  const float* xz    = dir ? xz_b  : xz_f;
  const float* h_in  = dir ? hb_in : hf_in;
  float*       h_out = dir ? hb_out : hf_out;
  float*       c_st  = dir ? cb     : cf;

  const int colBase = blockIdx.x * 16;                 // h index block
  const int rowBase = (blockIdx.y * 4 + wave) * 16;    // batch block
  const int col     = colBase + lm;

  const float*       hrow = h_in + (long)(rowBase + lm) * HIDDEN;
  const unsigned int lds0 = lds_addr32(&Bs[0][0]);

  // Stage one 64-row chunk of the 4-gate R panel into LDS buffer `buf`.
  // Layout: Bs[buf][g*1024 + r*16 + c], all transfers 16B aligned.
  auto stage = [&](int kbase, int buf) {
    const unsigned int base = lds0 + (unsigned)buf * (4u * KCHUNK * 16u * 4u);
    #pragma unroll
    for (int i = tid; i < (4 * KCHUNK * 16) / 4; i += 128) {  // 8 iters/thread
      const int e   = i * 4;
      const int g   = e >> 10;         // gate
      const int rem = e & 1023;
      const int r   = rem >> 4;        // row within chunk
      const int c   = rem & 15;        // col within tile
      async_copy_b128(base + (unsigned)e * 4u,
                      R + (long)(kbase + r) * GATES + g * HIDDEN + colBase + c);
    }
  };

  v8f acc0 = {}, acc1 = {}, acc2 = {}, acc3 = {};

  stage(0, 0);
  wait_async0();
  __syncthreads();

  const int nChunks = HIDDEN / KCHUNK;   // 8
  for (int ci = 0; ci < nChunks; ++ci) {
    const int buf = ci & 1;
    if (ci + 1 < nChunks) stage((ci + 1) * KCHUNK, buf ^ 1);

    const float* Bb    = &Bs[buf][0];
    const int    kbase = ci * KCHUNK;

    #pragma unroll 4
    for (int kk = 0; kk < KCHUNK; kk += 4) {
      const int kl = kk + hi * 2;            // local K row in chunk
      const int kg = kbase + kl;             // global K row

      v2f a; a.x = hrow[kg]; a.y = hrow[kg + 1];

      v2f b0; b0.x = Bb[0 * 1024 + kl * 16 + lm]; b0.y = Bb[0 * 1024 + (kl + 1) * 16 + lm];
      acc0 = wmma_f32x4(a, b0, acc0);
      v2f b1; b1.x = Bb[1 * 1024 + kl * 16 + lm]; b1.y = Bb[1 * 1024 + (kl + 1) * 16 + lm];
      acc1 = wmma_f32x4(a, b1, acc1);
      v2f b2; b2.x = Bb[2 * 1024 + kl * 16 + lm]; b2.y = Bb[2 * 1024 + (kl + 1) * 16 + lm];
      acc2 = wmma_f32x4(a, b2, acc2);
      v2f b3; b3.x = Bb[3 * 1024 + kl * 16 + lm]; b3.y = Bb[3 * 1024 + (kl + 1) * 16 + lm];
      acc3 = wmma_f32x4(a, b3, acc3);
    }

    wait_async0();      // next chunk landed (and our copies retired)
    __syncthreads();    // all waves done reading `buf` before it's reused
  }

  #pragma unroll
  for (int v = 0; v < 8; ++v) {
    const int  b    = rowBase + v + hi * 8;
    const long zrow = ((long)b * SEQT + t) * GATES;
    const long sidx = (long)b * HIDDEN + col;

    const float zi = acc0[v] + xz[zrow + 0 * HIDDEN + col];
    const float zf = acc1[v] + xz[zrow + 1 * HIDDEN + col];
    const float zg = acc2[v] + xz[zrow + 2 * HIDDEN + col];
    const float zo = acc3[v] + xz[zrow + 3 * HIDDEN + col];

    const float c_old = c_st[sidx];
    const float h_old = h_in[sidx];

    const float c_new = sigmoid_f(zf) * c_old + sigmoid_f(zi) * tanh_f(zg);
    const float h_new = sigmoid_f(zo) * tanh_f(c_new);

    const bool  m  = (src_ids[b * SEQT + t] != 0);
    const float h2 = m ? h_new : h_old;
    const float c2 = m ? c_new : c_old;

    h_out[sidx] = h2;
    c_st[sidx]  = c2;
    seq_out[((long)b * SEQT + t) * OUT2H + dir * HIDDEN + col] = h2;
  }
}

// ---------------------------------------------------------------------------
// Pack final states: enc_hid = [h_f | h_b], enc_cell = [c_f | c_b]
// ---------------------------------------------------------------------------
__global__ __launch_bounds__(256)
void finalize_kernel(const float* __restrict__ hf, const float* __restrict__ cf,
                     const float* __restrict__ hb, const float* __restrict__ cb,
                     float* __restrict__ enc_hid, float* __restrict__ enc_cell)
{
  int i = blockIdx.x * blockDim.x + threadIdx.x;
  if (i >= BATCH * HIDDEN) return;
  const int b = i / HIDDEN;
  const int j = i % HIDDEN;
  enc_hid [(long)b * OUT2H + j]          = hf[i];
  enc_hid [(long)b * OUT2H + HIDDEN + j] = hb[i];
  enc_cell[(long)b * OUT2H + j]          = cf[i];
  enc_cell[(long)b * OUT2H + HIDDEN + j] = cb[i];
}

// ---------------------------------------------------------------------------
extern "C" void kernel_launch(void* const* d_in, const int* in_sizes, int n_in,
                              void* d_out, int out_size, void* d_ws, size_t ws_size,
                              hipStream_t stream)
{
  (void)in_sizes; (void)n_in; (void)out_size; (void)ws_size;

  const int*   src_ids = (const int*)  d_in[0];
  const float* embed   = (const float*)d_in[1];
  const float* Kf      = (const float*)d_in[2];
  const float* Rf      = (const float*)d_in[3];
  const float* bf      = (const float*)d_in[4];
  const float* Kb      = (const float*)d_in[5];
  const float* Rb      = (const float*)d_in[6];
  const float* bb      = (const float*)d_in[7];

  float* out      = (float*)d_out;
  float* seq      = out;                                     // (B, T, 2H)
  float* enc_hid  = seq + (long)BATCH * SEQT * OUT2H;        // (B, 2H)
  float* enc_cell = enc_hid + (long)BATCH * OUT2H;           // (B, 2H)
  float* mask_out = enc_cell + (long)BATCH * OUT2H;          // (B, T)

  float* ws   = (float*)d_ws;
  float* xz_f = ws;                                          // (B*T, G)
  float* xz_b = xz_f + (long)BATCH * SEQT * GATES;
  float* hf0  = xz_b + (long)BATCH * SEQT * GATES;
  float* hf1  = hf0 + (long)BATCH * HIDDEN;
  float* cf   = hf1 + (long)BATCH * HIDDEN;
  float* hb0  = cf  + (long)BATCH * HIDDEN;
  float* hb1  = hb0 + (long)BATCH * HIDDEN;
  float* cb   = hb1 + (long)BATCH * HIDDEN;

  // 1) zero state + mask output
  init_kernel<<<(BATCH * HIDDEN + 255) / 256, 256, 0, stream>>>(
      src_ids, mask_out, hf0, hf1, cf, hb0, hb1, cb);

  // 2) xz = gather(embed) @ K + bias, both directions
  {
    dim3 grid(GATES / 16, (BATCH * SEQT / 16) / 4, 2);
    xz_gemm_kernel<<<grid, 128, 0, stream>>>(src_ids, embed, Kf, bf, Kb, bb,
                                             xz_f, xz_b);
  }

  // 3) 64 recurrent steps (fwd t = s, bwd t = 63 - s), h ping-pong buffered
  {
    dim3 grid(HIDDEN / 16, (BATCH / 16) / 4, 2);
    for (int s = 0; s < SEQT; ++s) {
      float* hf_in  = (s & 1) ? hf1 : hf0;
      float* hf_out = (s & 1) ? hf0 : hf1;
      float* hb_in  = (s & 1) ? hb1 : hb0;
      float* hb_out = (s & 1) ? hb0 : hb1;
      lstm_step_kernel<<<grid, 128, 0, stream>>>(
          xz_f, xz_b, Rf, Rb, src_ids,
          hf_in, hf_out, cf, hb_in, hb_out, cb, seq, s);
    }
  }

  // 4) final states: step 63 (odd) wrote into buffer 0
  finalize_kernel<<<(BATCH * HIDDEN + 255) / 256, 256, 0, stream>>>(
      hf0, cf, hb0, cb, enc_hid, enc_cell);
}